// GlobalModel_31748398252730
// MI455X (gfx1250) — compile-verified
//
#include <hip/hip_runtime.h>

// ---------------------------------------------------------------------------
// GlobalModel (graph-net global block) for MI455X / gfx1250, wave32.
//   mean_h = scatter_mean(x_h, batch_h, B)          [B,128]
//   mean_g = scatter_mean(x_g, batch_g, B)          [B,128]
//   Z      = [u | mean_h | mean_g]                  [B,384]
//   H      = leaky_relu(Z @ W1 + b1, 0.1)           [B,128]
//   out    = H @ W2 + b2                            [B,128]
// Memory-bound: 512 MB of node features -> ~22 us floor @ 23.3 TB/s.
// MLP done with v_wmma_f32_16x16x4_f32 (fp32 WMMA, 16x16 tiles, wave32).
// ---------------------------------------------------------------------------

#define N_FEAT 128       // n_h == n_g == n_u == 128
#define D_IN   384       // n_u + n_h + n_g

typedef __attribute__((ext_vector_type(2))) float v2f;
typedef __attribute__((ext_vector_type(8))) float v8f;

// ---------------- workspace zeroing ----------------
__global__ void zero_ws_kernel(float* __restrict__ ws, int n) {
  int i = blockIdx.x * blockDim.x + threadIdx.x;
  if (i < n) ws[i] = 0.0f;
}

// ---------------- per-segment counts (LDS histogram, few global atomics) ----
__global__ void count_kernel(const int* __restrict__ bh, const int* __restrict__ bg,
                             int nh, int ng, int B,
                             unsigned* __restrict__ cnt_h, unsigned* __restrict__ cnt_g) {
  __shared__ unsigned sbin[256];                  // 2 * B bins, B <= 128
  for (int i = threadIdx.x; i < 2 * B; i += blockDim.x) sbin[i] = 0u;
  __syncthreads();
  const int stride = gridDim.x * blockDim.x;
  for (int i = blockIdx.x * blockDim.x + threadIdx.x; i < nh; i += stride) {
    int s = bh[i];
    if ((unsigned)s < (unsigned)B) atomicAdd(&sbin[s], 1u);
  }
  for (int i = blockIdx.x * blockDim.x + threadIdx.x; i < ng; i += stride) {
    int s = bg[i];
    if ((unsigned)s < (unsigned)B) atomicAdd(&sbin[B + s], 1u);
  }
  __syncthreads();
  for (int i = threadIdx.x; i < 2 * B; i += blockDim.x) {
    unsigned v = sbin[i];
    if (v) {
      if (i < B) atomicAdd(&cnt_h[i], v);
      else       atomicAdd(&cnt_g[i - B], v);
    }
  }
}

// ---------------- sorted-segment streaming sum ----------------
// One wave loads one full 512-byte row per iteration (float4/lane, b128 loads).
// Sorted batch ids => register accumulation with atomic flush only on segment
// change (~1-2 flushes per thread per block).
__device__ __forceinline__ void flush4(float* __restrict__ sum, int seg, int col,
                                       const float4& acc, int B) {
  if (seg >= 0 && seg < B) {
    float* p = sum + (size_t)seg * N_FEAT + col;
    unsafeAtomicAdd(p + 0, acc.x);
    unsafeAtomicAdd(p + 1, acc.y);
    unsafeAtomicAdd(p + 2, acc.z);
    unsafeAtomicAdd(p + 3, acc.w);
  }
}

__global__ void scatter_sum_kernel(const float* __restrict__ xh, const int* __restrict__ bh,
                                   float* __restrict__ sumh, int nh,
                                   const float* __restrict__ xg, const int* __restrict__ bg,
                                   float* __restrict__ sumg, int ng,
                                   int B, int rowsPerBlock) {
  const float* x; const int* batch; float* sum; int n;
  if (blockIdx.y == 0) { x = xh; batch = bh; sum = sumh; n = nh; }
  else                 { x = xg; batch = bg; sum = sumg; n = ng; }

  const int lane = threadIdx.x & 31;      // 32 lanes x float4 = one 128-col row
  const int wv   = threadIdx.x >> 5;      // wave id within block (row stream)
  const int nw   = blockDim.x  >> 5;
  const int col  = lane * 4;

  const int rBeg = blockIdx.x * rowsPerBlock;
  const int rEnd = min(n, rBeg + rowsPerBlock);

  float4 acc = make_float4(0.f, 0.f, 0.f, 0.f);
  int curSeg = -1;
  for (int r = rBeg + wv; r < rEnd; r += nw) {
    const int seg = batch[r];             // wave-uniform (whole wave = one row)
    if (seg != curSeg) {
      flush4(sum, curSeg, col, acc, B);
      curSeg = seg;
      acc = make_float4(0.f, 0.f, 0.f, 0.f);
    }
    const float4 v = *reinterpret_cast<const float4*>(x + (size_t)r * N_FEAT + col);
    acc.x += v.x; acc.y += v.y; acc.z += v.z; acc.w += v.w;
  }
  flush4(sum, curSeg, col, acc, B);
}

// ---------------- Z = [u | sum_h/cnt_h | sum_g/cnt_g] ----------------
__global__ void finalize_kernel(const float* __restrict__ u,
                                const float* __restrict__ sum_h, const unsigned* __restrict__ cnt_h,
                                const float* __restrict__ sum_g, const unsigned* __restrict__ cnt_g,
                                float* __restrict__ Z, int B) {
  int i = blockIdx.x * blockDim.x + threadIdx.x;
  if (i >= B * D_IN) return;
  const int b = i / D_IN;
  const int c = i - b * D_IN;
  float v;
  if (c < N_FEAT) {
    v = u[b * N_FEAT + c];
  } else if (c < 2 * N_FEAT) {
    v = sum_h[b * N_FEAT + (c - N_FEAT)] / fmaxf((float)cnt_h[b], 1.0f);
  } else {
    v = sum_g[b * N_FEAT + (c - 2 * N_FEAT)] / fmaxf((float)cnt_g[b], 1.0f);
  }
  Z[i] = v;
}

// ---------------- fused 2-layer MLP via V_WMMA_F32_16X16X4_F32 -------------
// One block, 32 waves; wave w owns output tile (m0 = (w>>3)*16, n0 = (w&7)*16).
// fp32 A 16x4 layout (ISA 7.12.2): lanes 0-15 -> K0,K1 ; lanes 16-31 -> K2,K3.
// fp32 B 4x16 / C,D 16x16: rows striped across lanes within a VGPR.
// No divergent control flow anywhere -> EXEC all-1s as WMMA requires.
__global__ __launch_bounds__(1024) void mlp_wmma_kernel(
    const float* __restrict__ Z,    // [64][384]
    const float* __restrict__ W1,   // [384][128]
    const float* __restrict__ b1,   // [128]
    const float* __restrict__ W2,   // [128][128]
    const float* __restrict__ b2,   // [128]
    float* __restrict__ out) {      // [64][128]
  __shared__ float Hs[64 * N_FEAT];               // 32 KB

  const int tid  = threadIdx.x;
  const int lane = tid & 31;
  const int w    = tid >> 5;                      // 0..31
  const int m0   = (w >> 3) * 16;                 // 0,16,32,48
  const int n0   = (w & 7) * 16;                  // 0..112
  const int half = lane >> 4;                     // 0: K0/K1, 1: K2/K3
  const int l    = lane & 15;
  const int mA   = m0 + l;                        // A-matrix row for this lane

  // ---- GEMM1: H = leaky_relu(Z @ W1 + b1) ----
  v8f c = {};
  #pragma unroll 4
  for (int k = 0; k < D_IN; k += 4) {
    const int ka = k + 2 * half;
    const float2 af = *reinterpret_cast<const float2*>(Z + (size_t)mA * D_IN + ka);
    v2f a; a.x = af.x; a.y = af.y;
    v2f b;
    b.x = W1[(size_t)ka * N_FEAT + n0 + l];
    b.y = W1[(size_t)(ka + 1) * N_FEAT + n0 + l];
    c = __builtin_amdgcn_wmma_f32_16x16x4_f32(false, a, false, b, (short)0, c,
                                              false, false);
  }
  #pragma unroll
  for (int v = 0; v < 8; ++v) {
    const int M = m0 + v + 8 * half;              // C/D layout: lanes 16-31 -> M+8
    const int N = n0 + l;
    float val = c[v] + b1[N];
    val = (val > 0.0f) ? val : 0.1f * val;        // leaky_relu(0.1)
    Hs[M * N_FEAT + N] = val;
  }
  __syncthreads();

  // ---- GEMM2: out = H @ W2 + b2 ----
  v8f c2 = {};
  #pragma unroll 4
  for (int k = 0; k < N_FEAT; k += 4) {
    const int ka = k + 2 * half;
    const float2 af = *reinterpret_cast<const float2*>(&Hs[mA * N_FEAT + ka]);
    v2f a; a.x = af.x; a.y = af.y;
    v2f b;
    b.x = W2[(size_t)ka * N_FEAT + n0 + l];
    b.y = W2[(size_t)(ka + 1) * N_FEAT + n0 + l];
    c2 = __builtin_amdgcn_wmma_f32_16x16x4_f32(false, a, false, b, (short)0, c2,
                                               false, false);
  }
  #pragma unroll
  for (int v = 0; v < 8; ++v) {
    const int M = m0 + v + 8 * half;
    const int N = n0 + l;
    out[M * N_FEAT + N] = c2[v] + b2[N];
  }
}

// ---------------------------------------------------------------------------
extern "C" void kernel_launch(void* const* d_in, const int* in_sizes, int n_in,
                              void* d_out, int out_size, void* d_ws, size_t ws_size,
                              hipStream_t stream) {
  const float* x_h     = (const float*)d_in[0];
  const float* x_g     = (const float*)d_in[1];
  // d_in[2] edge_index, d_in[3] edge_attr: unused by the reference
  const float* u       = (const float*)d_in[4];
  const int*   batch_h = (const int*)d_in[5];
  const int*   batch_g = (const int*)d_in[6];
  const float* W1      = (const float*)d_in[7];
  const float* b1      = (const float*)d_in[8];
  const float* W2      = (const float*)d_in[9];
  const float* b2      = (const float*)d_in[10];

  const int B  = in_sizes[4] / N_FEAT;   // 64
  const int nh = in_sizes[5];            // 500000
  const int ng = in_sizes[6];            // 500000

  // workspace layout (floats): sum_h[B*128] | sum_g[B*128] | cnt_h[B] | cnt_g[B] | Z[B*384]
  float*    ws    = (float*)d_ws;
  float*    sum_h = ws;
  float*    sum_g = sum_h + (size_t)B * N_FEAT;
  unsigned* cnt_h = (unsigned*)(sum_g + (size_t)B * N_FEAT);
  unsigned* cnt_g = cnt_h + B;
  float*    Z     = (float*)(cnt_g + B);

  // 1) zero accumulators + counters
  const int zeroN = 2 * B * N_FEAT + 2 * B;
  zero_ws_kernel<<<(zeroN + 255) / 256, 256, 0, stream>>>(ws, zeroN);

  // 2) segment counts (LDS histogram)
  count_kernel<<<256, 256, 0, stream>>>(batch_h, batch_g, nh, ng, B, cnt_h, cnt_g);

  // 3) streaming sorted-segment sums (the 512 MB bandwidth-bound phase)
  const int rowsPerBlock = 2048;
  const int nmax = (nh > ng) ? nh : ng;
  dim3 sgrid((nmax + rowsPerBlock - 1) / rowsPerBlock, 2);
  scatter_sum_kernel<<<sgrid, 256, 0, stream>>>(x_h, batch_h, sum_h, nh,
                                                x_g, batch_g, sum_g, ng,
                                                B, rowsPerBlock);

  // 4) Z = [u | mean_h | mean_g]
  const int zn = B * D_IN;
  finalize_kernel<<<(zn + 255) / 256, 256, 0, stream>>>(u, sum_h, cnt_h,
                                                        sum_g, cnt_g, Z, B);

  // 5) fused WMMA MLP -> d_out  (B == 64 tiles onto 32 waves)
  mlp_wmma_kernel<<<1, 1024, 0, stream>>>(Z, W1, b1, W2, b2, (float*)d_out);
}